// DetectionLoss_45887430590739
// MI455X (gfx1250) — compile-verified
//
#include <hip/hip_runtime.h>
#include <hip/hip_bf16.h>
#include <math.h>

#define NUM_B 40
#define NUM_CLS 3

typedef float v2f __attribute__((ext_vector_type(2)));
typedef float v8f __attribute__((ext_vector_type(8)));

// ---- order-preserving float<->uint key (total order matches float compare) ----
__device__ __forceinline__ unsigned fkey(float f) {
    unsigned u = __float_as_uint(f);
    return (u & 0x80000000u) ? ~u : (u | 0x80000000u);
}
__device__ __forceinline__ float fval(unsigned k) {
    return __uint_as_float((k & 0x80000000u) ? (k & 0x7FFFFFFFu) : ~k);
}
// stable softplus == jnp.logaddexp(x, 0)
__device__ __forceinline__ float softplusf(float x) {
    return fmaxf(x, 0.f) + log1pf(expf(-fabsf(x)));
}

// deterministic tree reductions over a 256-thread block
__device__ __forceinline__ float blockReduceF(float v, float* red) {
    int tid = threadIdx.x;
    red[tid] = v; __syncthreads();
    for (int s = 128; s > 0; s >>= 1) {
        if (tid < s) red[tid] += red[tid + s];
        __syncthreads();
    }
    float r = red[0]; __syncthreads();
    return r;
}
__device__ __forceinline__ int blockReduceI(int v, int* red) {
    int tid = threadIdx.x;
    red[tid] = v; __syncthreads();
    for (int s = 128; s > 0; s >>= 1) {
        if (tid < s) red[tid] += red[tid + s];
        __syncthreads();
    }
    int r = red[0]; __syncthreads();
    return r;
}

// -------------------- Phase 1: per-anchor matching & losses --------------------
// grid = (A/256, 16), block = 256. One thread = one anchor.
__global__ __launch_bounds__(256) void p1_kernel(
    const float* __restrict__ pred, const float* __restrict__ anchors,
    const float* __restrict__ boxes, const int* __restrict__ labels,
    float* __restrict__ negloss, float* __restrict__ blockpart,
    int HW, int A, int negOff, int sOff)
{
    __shared__ float sb[NUM_B * 4];
    __shared__ float sarea[NUM_B];
    __shared__ int   slab[NUM_B];
    __shared__ float red[256];

    const int b = blockIdx.y;
    const int tid = threadIdx.x;

    if (tid < NUM_B * 4) sb[tid] = boxes[b * NUM_B * 4 + tid];
    __syncthreads();
    if (tid < NUM_B) {
        float x0 = sb[tid*4+0], y0 = sb[tid*4+1], x1 = sb[tid*4+2], y1 = sb[tid*4+3];
        sarea[tid] = (x1 - x0) * (y1 - y0);
        slab[tid]  = labels[b * NUM_B + tid];
    }
    __syncthreads();

    const int a = blockIdx.x * 256 + tid;
    float objp = 0.f, clsl = 0.f, locl = 0.f, posf = 0.f, negf = 0.f;

    if (a < A) {
        float4 an = reinterpret_cast<const float4*>(anchors)[a];
        float aw = an.z - an.x, ah = an.w - an.y;
        float areaA = aw * ah;

        // best IoU over 40 boxes; first-occurrence argmax like jnp.argmax
        float best = -1.f; int bi = 0;
        #pragma unroll 4
        for (int j = 0; j < NUM_B; ++j) {
            float bx0 = sb[j*4+0], by0 = sb[j*4+1], bx1 = sb[j*4+2], by1 = sb[j*4+3];
            float lx = fmaxf(an.x, bx0), ly = fmaxf(an.y, by0);
            float rx = fminf(an.z, bx1), ry = fminf(an.w, by1);
            float w  = fmaxf(rx - lx, 0.f), h = fmaxf(ry - ly, 0.f);
            float inter = w * h;
            float iou = inter / (areaA + sarea[j] - inter + 1e-9f);
            if (iou > best) { best = iou; bi = j; }
        }
        bool pos = best >= 0.5f;
        bool neg = best <  0.3f;

        // channels: anchor a = (hw)*3 + s ; features j use channel s*8+j (stride HW)
        int hw = a / 3;
        int sa = a - hw * 3;
        size_t base = ((size_t)b * 24 + (size_t)sa * 8) * (size_t)HW + (size_t)hw;
        float p0 = pred[base + 0*(size_t)HW];
        float p1 = pred[base + 1*(size_t)HW];
        float p2 = pred[base + 2*(size_t)HW];
        float p3 = pred[base + 3*(size_t)HW];
        float p4 = pred[base + 4*(size_t)HW];
        float p5 = pred[base + 5*(size_t)HW];
        float p6 = pred[base + 6*(size_t)HW];
        float p7 = pred[base + 7*(size_t)HW];

        float sp = softplusf(p4);

        if (pos) {
            posf = 1.f;
            objp = sp - p4;                        // softplus(x) - x*1
            // classification CE (log_softmax over 3 logits)
            float m  = fmaxf(p5, fmaxf(p6, p7));
            float lse = m + logf(expf(p5 - m) + expf(p6 - m) + expf(p7 - m));
            int lbl = slab[bi];
            int tgt = min(max(lbl - 1, 0), NUM_CLS - 1);
            float ct = (tgt == 0) ? p5 : ((tgt == 1) ? p6 : p7);
            clsl = lse - ct;
            // box regression (smooth L1 on encoded deltas)
            float bx0 = sb[bi*4+0], by0 = sb[bi*4+1], bx1 = sb[bi*4+2], by1 = sb[bi*4+3];
            float gw = bx1 - bx0, gh = by1 - by0;
            float gcx = bx0 + 0.5f * gw, gcy = by0 + 0.5f * gh;
            float acx = an.x + 0.5f * aw, acy = an.y + 0.5f * ah;
            float e0 = (gcx - acx) / aw;
            float e1 = (gcy - acy) / ah;
            float e2 = logf(gw / aw);
            float e3 = logf(gh / ah);
            float d0 = p0 - e0, d1 = p1 - e1, d2 = p2 - e2, d3 = p3 - e3;
            float a0 = fabsf(d0), a1 = fabsf(d1), a2 = fabsf(d2), a3 = fabsf(d3);
            locl  = (a0 < 1.f ? 0.5f * d0 * d0 : a0 - 0.5f);
            locl += (a1 < 1.f ? 0.5f * d1 * d1 : a1 - 0.5f);
            locl += (a2 < 1.f ? 0.5f * d2 * d2 : a2 - 0.5f);
            locl += (a3 < 1.f ? 0.5f * d3 * d3 : a3 - 0.5f);
        }
        if (neg) negf = 1.f;
        // negative candidate loss array (softplus for negs, -1 sentinel otherwise)
        negloss[negOff + b * A + a] = neg ? sp : -1.0f;
    }

    float rObj = blockReduceF(objp, red);
    float rCls = blockReduceF(clsl, red);
    float rLoc = blockReduceF(locl, red);
    float rPos = blockReduceF(posf, red);
    float rNeg = blockReduceF(negf, red);
    if (tid == 0) {
        float* bp = blockpart + (size_t)((b * 252 + sOff + blockIdx.x)) * 8;
        bp[0] = rObj; bp[1] = rCls; bp[2] = rLoc; bp[3] = rPos; bp[4] = rNeg;
    }
}

// ------------- Phase 2: per-(batch,scale) fold + exact top-k mining -------------
// grid = 48 (b*3+s), block = 256. Radix threshold search on monotone uint keys.
__global__ __launch_bounds__(256) void p2_kernel(
    const float* __restrict__ negloss, const float* __restrict__ blockpart,
    float* __restrict__ res)
{
    __shared__ float fred[256];
    __shared__ int   ired[256];

    const int bs = blockIdx.x;
    const int b = bs / 3, s = bs - b * 3;
    const int A_[3]   = {49152, 12288, 3072};
    const int nb_[3]  = {192, 48, 12};
    const int off_[3] = {0, 786432, 983040};
    const int so_[3]  = {0, 192, 240};
    const int A  = A_[s];
    const int nb = nb_[s];
    const float* arr = negloss + off_[s] + (size_t)b * A;
    const int tid = threadIdx.x;

    // fold per-block partials (deterministic)
    float q[5];
    for (int j = 0; j < 5; ++j) {
        float v = 0.f;
        for (int i = tid; i < nb; i += 256)
            v += blockpart[(size_t)(b * 252 + so_[s] + i) * 8 + j];
        q[j] = blockReduceF(v, fred);
    }
    const int posI = (int)(q[3] + 0.5f);
    const int negI = (int)(q[4] + 0.5f);
    const int k = min(negI, 3 * max(posI, 1));

    float selSum = 0.f;
    if (k > 0) {
        // find key of k-th largest value: 32-bit radix descent
        unsigned cand = 0u;
        for (int bit = 31; bit >= 0; --bit) {
            unsigned c2 = cand | (1u << bit);
            int cnt = 0;
            for (int i = tid; i < A; i += 256)
                cnt += (fkey(arr[i]) >= c2) ? 1 : 0;
            cnt = blockReduceI(cnt, ired);
            if (cnt >= k) cand = c2;       // uniform across block
        }
        // sum strictly-above-threshold + ties at threshold (value-exact)
        float sg = 0.f; int cg = 0;
        for (int i = tid; i < A; i += 256) {
            float v = arr[i];
            if (fkey(v) > cand) { sg += v; cg++; }
        }
        sg = blockReduceF(sg, fred);
        cg = blockReduceI(cg, ired);
        selSum = sg + (float)(k - cg) * fval(cand);
    }

    if (tid == 0) {
        float* r = res + bs * 8;
        r[0] = q[0];        // pos obj loss
        r[1] = q[1];        // cls loss
        r[2] = q[2];        // loc loss
        r[3] = q[3];        // pos count
        r[4] = (float)k;    // selected neg count (== sel_neg.sum())
        r[5] = selSum;      // selected neg obj loss
        r[6] = q[4];
        r[7] = 0.f;
    }
}

// ---------------- Phase 3: final 48-entry sums via f32 WMMA ----------------
// One wave32. Sum-of-column-sums trick: D = ones(16x4) x B(4x16) gives column
// sums; summing the 16 columns yields the total of all 64 B elements
// independent of the exact lane->(k,n) mapping.
__global__ __launch_bounds__(32) void p3_kernel(
    const float* __restrict__ res, float* __restrict__ out)
{
    const int l = threadIdx.x;
    float sums[6];
    #pragma unroll
    for (int q = 0; q < 6; ++q) {
        float e0 = res[l * 8 + q];
        float e1 = (l < 16) ? res[(l + 32) * 8 + q] : 0.f;   // entries 32..47
#if defined(__gfx1250__) && __has_builtin(__builtin_amdgcn_wmma_f32_16x16x4_f32)
        v2f av; av.x = 1.f; av.y = 1.f;
        v2f bv; bv.x = e0;  bv.y = e1;
        v8f c = {0.f, 0.f, 0.f, 0.f, 0.f, 0.f, 0.f, 0.f};
        v8f d = __builtin_amdgcn_wmma_f32_16x16x4_f32(
            false, av, false, bv, (short)0, c, false, false);
        float cs = d[0];                       // column sum, replicated per 16-group
        cs += __shfl_xor(cs, 1, 32);
        cs += __shfl_xor(cs, 2, 32);
        cs += __shfl_xor(cs, 4, 32);
        cs += __shfl_xor(cs, 8, 32);
        sums[q] = cs;                          // total in every lane
#else
        float t = e0 + e1;
        for (int m = 1; m < 32; m <<= 1) t += __shfl_xor(t, m, 32);
        sums[q] = t;
#endif
    }
    if (l == 0) {
        float sObj = sums[0], sCls = sums[1], sLoc = sums[2];
        float sPos = sums[3], sK = sums[4], sSel = sums[5];
        float norm = fmaxf(sPos, 1.f);
        float lo = (sObj + sSel) / norm;
        float lc = sCls / norm;
        float ll = sLoc / norm;
        out[0] = lo;
        out[1] = lc;
        out[2] = ll;
        out[3] = lo + lc + 2.f * ll;
        out[4] = sPos;
        out[5] = sK;
    }
}

extern "C" void kernel_launch(void* const* d_in, const int* in_sizes, int n_in,
                              void* d_out, int out_size, void* d_ws, size_t ws_size,
                              hipStream_t stream) {
    const float* pred0    = (const float*)d_in[0];
    const float* pred1    = (const float*)d_in[1];
    const float* pred2    = (const float*)d_in[2];
    const float* anchors0 = (const float*)d_in[3];
    const float* anchors1 = (const float*)d_in[4];
    const float* anchors2 = (const float*)d_in[5];
    const float* boxes    = (const float*)d_in[6];
    const int*   labels   = (const int*)d_in[7];

    float* ws        = (float*)d_ws;
    float* negloss   = ws;                       // 1,032,192 floats
    float* blockpart = ws + 1032192;             // 4032 * 8 floats
    float* res       = blockpart + 4032 * 8;     // 48 * 8 floats

    // scale 0: 128x128, A=49152, 192 blocks/batch
    p1_kernel<<<dim3(192, 16), 256, 0, stream>>>(
        pred0, anchors0, boxes, labels, negloss, blockpart, 16384, 49152, 0, 0);
    // scale 1: 64x64, A=12288, 48 blocks/batch
    p1_kernel<<<dim3(48, 16), 256, 0, stream>>>(
        pred1, anchors1, boxes, labels, negloss, blockpart, 4096, 12288, 786432, 192);
    // scale 2: 32x32, A=3072, 12 blocks/batch
    p1_kernel<<<dim3(12, 16), 256, 0, stream>>>(
        pred2, anchors2, boxes, labels, negloss, blockpart, 1024, 3072, 983040, 240);

    p2_kernel<<<48, 256, 0, stream>>>(negloss, blockpart, res);
    p3_kernel<<<1, 32, 0, stream>>>(res, (float*)d_out);
}